// H2GCN_30116310680317
// MI455X (gfx1250) — compile-verified
//
#include <hip/hip_runtime.h>

// ---------------- problem constants (match reference) ----------------
#define NROWS 10000
#define IN_C  2048
#define HID   256
#define OUT_C 256
#define E1    160000
#define E2    320000

// LDS layout for the 64K x 64N W tile: element (k,n) at dword
//   (k>>1)*PSTRIDE + n*2 + (k&1)
// so a lane's B fragment {K=kb, K=kb+1} is one aligned ds_load_b64.
// PSTRIDE = 128 + 32 pad: the two lane-halves (K-pairs p and p+1) are
// offset by 160 = 32 (mod 64 banks) -> disjoint bank sets, no conflicts.
#define PSTRIDE 160
#define LDS_CHUNK (32 * PSTRIDE)   // 32 K-pairs per 64-deep chunk

typedef float v2f __attribute__((ext_vector_type(2)));
typedef float v8f __attribute__((ext_vector_type(8)));

// ---------------------------------------------------------------------
// Zero-fill (h1/h2 accumulators live in workspace; harness poisons ws)
// ---------------------------------------------------------------------
__global__ void h2gcn_fill_zero(float* __restrict__ p, int n) {
    int i = blockIdx.x * blockDim.x + threadIdx.x;
    if (i < n) p[i] = 0.0f;
}

// ---------------------------------------------------------------------
// GEMM1: h0[N, HID] = x[N, IN_C] @ W1[IN_C, HID]
//
// Block = 256 threads (8 waves). Wave w owns a 16x64 output strip:
// rows m0=(blockIdx.x*8+w)*16, cols nb=blockIdx.y*64 (4 x v8f accums;
// each A fragment feeds 4 WMMAs). K staged 64-deep, double-buffered,
// K-pair-interleaved in LDS (see PSTRIDE above). A fragments (v2f per
// lane) stream from global; x is L2-resident across the 4 N-tiles.
//
// f32 WMMA A-layout (16x4): lanes 0-15 hold M=lane, K={0,1}; lanes
// 16-31 hold M=lane-16, K={2,3}. B (4x16): VGPR0 = K{0|2} row striped
// over N=lane%16, VGPR1 = K{1|3}.
// ---------------------------------------------------------------------
__global__ void h2gcn_gemm1(const float* __restrict__ x,
                            const float* __restrict__ W1,
                            float* __restrict__ h0) {
    __shared__ float lB[2][LDS_CHUNK];
    const int w    = threadIdx.x >> 5;
    const int lane = threadIdx.x & 31;
    const int m0   = (blockIdx.x * 8 + w) * 16;
    const int nb   = blockIdx.y * 64;
    const bool active = (m0 < NROWS);

    v8f acc[4];
    #pragma unroll
    for (int nt = 0; nt < 4; ++nt) acc[nt] = 0.0f;

    int aRow = m0 + (lane & 15);
    if (aRow >= NROWS) aRow = NROWS - 1;   // keep loads legal; results discarded
    const float* aBase = x + (size_t)aRow * IN_C + ((lane >> 4) << 1);

    // staging: 512 (K-pair, n4) units per chunk, 2 per thread.
    // unit u: p = u>>4 (K-pair 0..31), n4 = u&15 (4-wide N group).
    float4 sA[2], sB[2];
    auto loadStage = [&](int k0) {
        #pragma unroll
        for (int e = 0; e < 2; ++e) {
            const int u  = threadIdx.x + 256 * e;
            const int p  = u >> 4;
            const int n4 = (u & 15) << 2;
            const float* g = &W1[(size_t)(k0 + 2 * p) * HID + nb + n4];
            sA[e] = *(const float4*)g;
            sB[e] = *(const float4*)(g + HID);
        }
    };
    auto storeStage = [&](int buf) {
        #pragma unroll
        for (int e = 0; e < 2; ++e) {
            const int u  = threadIdx.x + 256 * e;
            const int p  = u >> 4;
            const int n4 = (u & 15) << 2;
            float* d = &lB[buf][p * PSTRIDE + n4 * 2];
            const float a0[4] = {sA[e].x, sA[e].y, sA[e].z, sA[e].w};
            const float b0[4] = {sB[e].x, sB[e].y, sB[e].z, sB[e].w};
            #pragma unroll
            for (int i = 0; i < 4; ++i) {
                v2f pr; pr.x = a0[i]; pr.y = b0[i];
                *(v2f*)(d + 2 * i) = pr;
            }
        }
    };

    constexpr int NCHUNK = IN_C / 64;      // 32
    loadStage(0);
    for (int c = 0; c < NCHUNK; ++c) {
        const int buf = c & 1;
        storeStage(buf);
        __syncthreads();
        if (c + 1 < NCHUNK) loadStage((c + 1) * 64);

        const float* ap = aBase + c * 64;
        #pragma unroll
        for (int kk = 0; kk < 64; kk += 4) {
            const v2f a = *(const v2f*)(ap + kk);
            const int p = (kk >> 1) + (lane >> 4);          // K-pair index
            const float* lb = &lB[buf][p * PSTRIDE + (lane & 15) * 2];
            #pragma unroll
            for (int nt = 0; nt < 4; ++nt) {
                const v2f b = *(const v2f*)(lb + nt * 32);  // 16 cols * 2
                acc[nt] = __builtin_amdgcn_wmma_f32_16x16x4_f32(
                    false, a, false, b, (short)0, acc[nt], false, false);
            }
        }
        __syncthreads();   // protect buf reuse two iterations later
    }

    if (active) {
        const int rbase = m0 + ((lane >> 4) << 3);
        #pragma unroll
        for (int nt = 0; nt < 4; ++nt) {
            const int col = nb + nt * 16 + (lane & 15);
            #pragma unroll
            for (int v = 0; v < 8; ++v)
                h0[(size_t)(rbase + v) * HID + col] = acc[nt][v];
        }
    }
}

// ---------------------------------------------------------------------
// SpMM: out[row] += val * h0[col]  (edge-parallel, one wave per edge)
// h0 is L2-resident (10 MB); scatter via global_atomic_add_f32.
// ---------------------------------------------------------------------
__global__ void h2gcn_spmm(const int* __restrict__ rows,
                           const int* __restrict__ cols,
                           const float* __restrict__ vals,
                           const float* __restrict__ h0,
                           float* __restrict__ out, int E) {
    const int edge = blockIdx.x * (blockDim.x >> 5) + (threadIdx.x >> 5);
    const int lane = threadIdx.x & 31;
    if (edge >= E) return;
    const int   r = rows[edge];
    const int   c = cols[edge];
    const float v = vals[edge];
    const float4* src = (const float4*)(h0 + (size_t)c * HID);
    float* dst = out + (size_t)r * HID;
    #pragma unroll
    for (int i = 0; i < 2; ++i) {
        const int j = lane + 32 * i;         // float4 index, 64 per row
        const float4 s = src[j];
        atomicAdd(dst + 4 * j + 0, v * s.x);
        atomicAdd(dst + 4 * j + 1, v * s.y);
        atomicAdd(dst + 4 * j + 2, v * s.z);
        atomicAdd(dst + 4 * j + 3, v * s.w);
    }
}

// ---------------------------------------------------------------------
// GEMM2: out[N, OUT_C] = [h0|h1|h2] @ W_out[3*HID, OUT_C] + b_out
// Same 16x64-per-wave WMMA tiling; the concat is walked as a flattened
// 12-chunk double-buffered pipeline (3 parts x 4 K-chunks of 64).
// Accumulators start from the bias.
// ---------------------------------------------------------------------
__global__ void h2gcn_gemm2(const float* __restrict__ h0,
                            const float* __restrict__ h1,
                            const float* __restrict__ h2,
                            const float* __restrict__ Wout,
                            const float* __restrict__ bout,
                            float* __restrict__ out) {
    __shared__ float lB[2][LDS_CHUNK];
    const int w    = threadIdx.x >> 5;
    const int lane = threadIdx.x & 31;
    const int m0   = (blockIdx.x * 8 + w) * 16;
    const int nb   = blockIdx.y * 64;
    const bool active = (m0 < NROWS);

    v8f acc[4];
    #pragma unroll
    for (int nt = 0; nt < 4; ++nt) {
        const float b = bout[nb + nt * 16 + (lane & 15)];
        #pragma unroll
        for (int v = 0; v < 8; ++v) acc[nt][v] = b;
    }

    int aRow = m0 + (lane & 15);
    if (aRow >= NROWS) aRow = NROWS - 1;
    const size_t aOff = (size_t)aRow * HID + ((lane >> 4) << 1);

    float4 sA[2], sB[2];
    auto loadStage = [&](int c) {          // c = flattened chunk 0..11
        const int part = c >> 2;
        const int k0   = (c & 3) * 64;
        const float* wp = Wout + (size_t)part * HID * OUT_C;
        #pragma unroll
        for (int e = 0; e < 2; ++e) {
            const int u  = threadIdx.x + 256 * e;
            const int p  = u >> 4;
            const int n4 = (u & 15) << 2;
            const float* g = &wp[(size_t)(k0 + 2 * p) * OUT_C + nb + n4];
            sA[e] = *(const float4*)g;
            sB[e] = *(const float4*)(g + OUT_C);
        }
    };
    auto storeStage = [&](int buf) {
        #pragma unroll
        for (int e = 0; e < 2; ++e) {
            const int u  = threadIdx.x + 256 * e;
            const int p  = u >> 4;
            const int n4 = (u & 15) << 2;
            float* d = &lB[buf][p * PSTRIDE + n4 * 2];
            const float a0[4] = {sA[e].x, sA[e].y, sA[e].z, sA[e].w};
            const float b0[4] = {sB[e].x, sB[e].y, sB[e].z, sB[e].w};
            #pragma unroll
            for (int i = 0; i < 4; ++i) {
                v2f pr; pr.x = a0[i]; pr.y = b0[i];
                *(v2f*)(d + 2 * i) = pr;
            }
        }
    };

    auto srcOf = [&](int p) { return p == 0 ? h0 : (p == 1 ? h1 : h2); };

    constexpr int NCHUNK = 12;             // 3 parts x (256/64)
    loadStage(0);
    for (int c = 0; c < NCHUNK; ++c) {
        const int buf = c & 1;
        storeStage(buf);
        __syncthreads();
        if (c + 1 < NCHUNK) loadStage(c + 1);

        const float* ap = srcOf(c >> 2) + aOff + (c & 3) * 64;
        #pragma unroll
        for (int kk = 0; kk < 64; kk += 4) {
            const v2f a = *(const v2f*)(ap + kk);
            const int p = (kk >> 1) + (lane >> 4);
            const float* lb = &lB[buf][p * PSTRIDE + (lane & 15) * 2];
            #pragma unroll
            for (int nt = 0; nt < 4; ++nt) {
                const v2f b = *(const v2f*)(lb + nt * 32);
                acc[nt] = __builtin_amdgcn_wmma_f32_16x16x4_f32(
                    false, a, false, b, (short)0, acc[nt], false, false);
            }
        }
        __syncthreads();
    }

    if (active) {
        const int rbase = m0 + ((lane >> 4) << 3);
        #pragma unroll
        for (int nt = 0; nt < 4; ++nt) {
            const int col = nb + nt * 16 + (lane & 15);
            #pragma unroll
            for (int v = 0; v < 8; ++v)
                out[(size_t)(rbase + v) * OUT_C + col] = acc[nt][v];
        }
    }
}

// ---------------------------------------------------------------------
extern "C" void kernel_launch(void* const* d_in, const int* in_sizes, int n_in,
                              void* d_out, int out_size, void* d_ws, size_t ws_size,
                              hipStream_t stream) {
    (void)in_sizes; (void)n_in; (void)out_size; (void)ws_size;

    const float* x    = (const float*)d_in[0];
    const int*   a1r  = (const int*)  d_in[1];
    const int*   a1c  = (const int*)  d_in[2];
    const float* a1v  = (const float*)d_in[3];
    const int*   a2r  = (const int*)  d_in[4];
    const int*   a2c  = (const int*)  d_in[5];
    const float* a2v  = (const float*)d_in[6];
    const float* W1   = (const float*)d_in[7];
    const float* Wout = (const float*)d_in[8];
    const float* bout = (const float*)d_in[9];
    float* out = (float*)d_out;

    float* h0 = (float*)d_ws;
    float* h1 = h0 + (size_t)NROWS * HID;
    float* h2 = h1 + (size_t)NROWS * HID;

    // zero h1 and h2 (contiguous in ws)
    {
        const int n = 2 * NROWS * HID;
        h2gcn_fill_zero<<<dim3((n + 255) / 256), dim3(256), 0, stream>>>(h1, n);
    }

    // h0 = x @ W1   (625 M-tiles of 16 rows, 8 per block; 4 N-tiles of 64)
    h2gcn_gemm1<<<dim3(79, 4), dim3(256), 0, stream>>>(x, W1, h0);

    // h1 = A1 @ h0, h2 = A2 @ h0   (one wave per edge, 8 edges/block)
    h2gcn_spmm<<<dim3((E1 + 7) / 8), dim3(256), 0, stream>>>(a1r, a1c, a1v, h0, h1, E1);
    h2gcn_spmm<<<dim3((E2 + 7) / 8), dim3(256), 0, stream>>>(a2r, a2c, a2v, h0, h2, E2);

    // out = [h0|h1|h2] @ W_out + b_out
    h2gcn_gemm2<<<dim3(79, 4), dim3(256), 0, stream>>>(h0, h1, h2, Wout, bout, out);
}